// TheoreticalModel_22333829939614
// MI455X (gfx1250) — compile-verified
//
#include <hip/hip_runtime.h>
#include <math.h>

typedef float v8f __attribute__((ext_vector_type(8)));
typedef float v2f __attribute__((ext_vector_type(2)));

#define BB      4096
#define DIMQ    512
#define DIMZ    64
#define HIDV    512
#define SEQ     128
#define T_OFF_STEPS 19
#define T_ON_STEPS  10

// ---------------------------------------------------------------------------
// Offline rollout: 19 steps of  q' = q + 0.1 p ; p' = (p - 0.01 q)*0.99
// Writes qf to qf_out, and K+Vquad terms for columns 0 (t=0) and 1 (t=f)
// of the offline_F output.
// ---------------------------------------------------------------------------
__global__ __launch_bounds__(512)
void dyn_off_kernel(const float* __restrict__ q0, const float* __restrict__ p0,
                    float* __restrict__ qf_out, float* __restrict__ F)
{
    const int b = blockIdx.x;
    const int d = threadIdx.x;
    float q = q0[(size_t)b * DIMQ + d];
    float p = p0[(size_t)b * DIMQ + d];
    float sp0 = p * p, sq0 = q * q;
#pragma unroll
    for (int t = 0; t < T_OFF_STEPS; ++t) {
        float qn = q + 0.1f * p;
        p = (p - 0.01f * q) * 0.99f;
        q = qn;
    }
    qf_out[(size_t)b * DIMQ + d] = q;
    float spf = p * p, sqf = q * q;

    __shared__ float red[16][4];
    const int lane = threadIdx.x & 31, wv = threadIdx.x >> 5;
#pragma unroll
    for (int m = 16; m; m >>= 1) {
        sp0 += __shfl_xor(sp0, m, 32);
        sq0 += __shfl_xor(sq0, m, 32);
        spf += __shfl_xor(spf, m, 32);
        sqf += __shfl_xor(sqf, m, 32);
    }
    if (lane == 0) { red[wv][0] = sp0; red[wv][1] = sq0; red[wv][2] = spf; red[wv][3] = sqf; }
    __syncthreads();
    if (threadIdx.x == 0) {
        float a = 0.f, c = 0.f, e = 0.f, g = 0.f;
#pragma unroll
        for (int w = 0; w < 16; ++w) { a += red[w][0]; c += red[w][1]; e += red[w][2]; g += red[w][3]; }
        F[(size_t)b * 2 + 0] = 0.5f * a + 0.05f * c;   // K(p0) + Vquad(q0)
        F[(size_t)b * 2 + 1] = 0.5f * e + 0.05f * g;   // K(pf) + Vquad(qf)
    }
}

// ---------------------------------------------------------------------------
// Online rollout: 10 steps of p' = (p - 0.01 q + 0.01 u)*0.99 ; q' = q + 0.1 p'
// ---------------------------------------------------------------------------
__global__ __launch_bounds__(512)
void dyn_on_kernel(const float* __restrict__ q0, const float* __restrict__ p0,
                   const float* __restrict__ u_seq,
                   float* __restrict__ qf_out, float* __restrict__ F)
{
    const int b = blockIdx.x;
    const int d = threadIdx.x;
    float q = q0[(size_t)b * DIMQ + d];
    float p = p0[(size_t)b * DIMQ + d];
    float sp0 = p * p, sq0 = q * q;
    const float* ub = u_seq + (size_t)b * SEQ * DIMQ + d;
#pragma unroll
    for (int t = 0; t < T_ON_STEPS; ++t) {
        float u = ub[(size_t)t * DIMQ];
        p = (p - 0.01f * q + 0.01f * u) * 0.99f;
        q = q + 0.1f * p;
    }
    qf_out[(size_t)b * DIMQ + d] = q;
    float spf = p * p, sqf = q * q;

    __shared__ float red[16][4];
    const int lane = threadIdx.x & 31, wv = threadIdx.x >> 5;
#pragma unroll
    for (int m = 16; m; m >>= 1) {
        sp0 += __shfl_xor(sp0, m, 32);
        sq0 += __shfl_xor(sq0, m, 32);
        spf += __shfl_xor(spf, m, 32);
        sqf += __shfl_xor(sqf, m, 32);
    }
    if (lane == 0) { red[wv][0] = sp0; red[wv][1] = sq0; red[wv][2] = spf; red[wv][3] = sqf; }
    __syncthreads();
    if (threadIdx.x == 0) {
        float a = 0.f, c = 0.f, e = 0.f, g = 0.f;
#pragma unroll
        for (int w = 0; w < 16; ++w) { a += red[w][0]; c += red[w][1]; e += red[w][2]; g += red[w][3]; }
        F[(size_t)b * 2 + 0] = 0.5f * a + 0.05f * c;
        F[(size_t)b * 2 + 1] = 0.5f * e + 0.05f * g;
    }
}

// ---------------------------------------------------------------------------
// V_net rows: out[row*stride] += tanh(Q@W1[:512] + z@W1[512:] + b1) @ W2 + b2
// One block = 16 batch rows. 8 waves, each wave does 4 of the 32 N-tiles.
// f32 WMMA 16x16x4, K-loop of 128 steps over DIMQ=512.
//   A (16x4 f32): lane l: M = l%16; VGPR0/1 = K = 2*(l/16) + {0,1}
//   B (4x16 f32): lane l: N = l%16; VGPR0/1 = K = 2*(l/16) + {0,1}
//   C (16x16 f32): VGPR r, lane l: M = r + 8*(l/16), N = l%16
// ---------------------------------------------------------------------------
__global__ __launch_bounds__(256)
void vnet_rows_kernel(const float* __restrict__ Q,   // [4096,512]
                      const float* __restrict__ W1,  // [576,512]
                      const float* __restrict__ b1,  // [512]
                      const float* __restrict__ W2,  // [512]
                      const float* __restrict__ z,   // [64]
                      const float* __restrict__ b2,  // [1]
                      float* __restrict__ outF,      // column base in output
                      int out_stride)
{
    const int rowbase = blockIdx.x * 16;
    const int wave = threadIdx.x >> 5;   // 0..7
    const int lane = threadIdx.x & 31;
    const int half = lane >> 4;          // 0 | 1
    const int l16  = lane & 15;

    __shared__ float red[8][16];

    float partial[8];
#pragma unroll
    for (int r = 0; r < 8; ++r) partial[r] = 0.f;

    const float* qrow = Q + (size_t)(rowbase + l16) * DIMQ + 2 * half;

    for (int nt = wave * 4; nt < wave * 4 + 4; ++nt) {
        const int n = nt * 16 + l16;
        const float* wcol = W1 + n + (size_t)(2 * half) * HIDV;
        v8f c = {};
#pragma unroll 4
        for (int k = 0; k < DIMQ; k += 4) {
            v2f a, bfrag;
            a.x = qrow[k];
            a.y = qrow[k + 1];
            bfrag.x = wcol[(size_t)k * HIDV];
            bfrag.y = wcol[(size_t)k * HIDV + HIDV];
            c = __builtin_amdgcn_wmma_f32_16x16x4_f32(
                    /*neg_a=*/false, a, /*neg_b=*/false, bfrag,
                    /*c_mod=*/(short)0, c, /*reuse_a=*/false, /*reuse_b=*/false);
        }
        // per-column bias: b1[n] + z @ W1[512:576, n]  (batch-constant)
        float zb = b1[n];
#pragma unroll 8
        for (int j = 0; j < DIMZ; ++j)
            zb += z[j] * W1[(size_t)(DIMQ + j) * HIDV + n];
        const float w2n = W2[n];
#pragma unroll
        for (int r = 0; r < 8; ++r)
            partial[r] += tanhf(c[r] + zb) * w2n;
    }

    // reduce across the 16 lanes of each half (xor masks stay within group)
#pragma unroll
    for (int r = 0; r < 8; ++r) {
        float v = partial[r];
        v += __shfl_xor(v, 1, 32);
        v += __shfl_xor(v, 2, 32);
        v += __shfl_xor(v, 4, 32);
        v += __shfl_xor(v, 8, 32);
        partial[r] = v;
    }
    if (l16 == 0) {   // lane 0 -> rows 0..7, lane 16 -> rows 8..15
#pragma unroll
        for (int r = 0; r < 8; ++r) red[wave][half * 8 + r] = partial[r];
    }
    __syncthreads();
    if (threadIdx.x < 16) {
        const int m = threadIdx.x;
        float v = 0.f;
#pragma unroll
        for (int w = 0; w < 8; ++w) v += red[w][m];
        const size_t row = rowbase + m;
        outF[row * out_stride] += v + b2[0];
    }
}

// ---------------------------------------------------------------------------
extern "C" void kernel_launch(void* const* d_in, const int* in_sizes, int n_in,
                              void* d_out, int out_size, void* d_ws, size_t ws_size,
                              hipStream_t stream) {
    const float* u_seq  = (const float*)d_in[0];
    const float* q0_off = (const float*)d_in[1];
    const float* p0_off = (const float*)d_in[2];
    const float* q0_on  = (const float*)d_in[3];
    const float* p0_on  = (const float*)d_in[4];
    const float* z      = (const float*)d_in[5];
    const float* W1     = (const float*)d_in[6];
    const float* b1     = (const float*)d_in[7];
    const float* W2     = (const float*)d_in[8];
    const float* b2     = (const float*)d_in[9];

    float* out   = (float*)d_out;
    float* offF  = out;                 // [4096,2]
    float* onF   = out + BB * 2;        // [4096,2]
    float* qtraj = out + BB * 4;        // [4096,512] == qf_on

    const dim3 dynGrid(BB), dynBlk(512);
    const dim3 gGrid(BB / 16), gBlk(256);
    const size_t needQ = (size_t)BB * DIMQ * sizeof(float);

    if (ws_size >= needQ) {
        float* qf_off = (float*)d_ws;
        dyn_off_kernel<<<dynGrid, dynBlk, 0, stream>>>(q0_off, p0_off, qf_off, offF);
        dyn_on_kernel <<<dynGrid, dynBlk, 0, stream>>>(q0_on,  p0_on, u_seq, qtraj, onF);
        vnet_rows_kernel<<<gGrid, gBlk, 0, stream>>>(q0_off, W1, b1, W2, z, b2, offF + 0, 2);
        vnet_rows_kernel<<<gGrid, gBlk, 0, stream>>>(qf_off, W1, b1, W2, z, b2, offF + 1, 2);
        vnet_rows_kernel<<<gGrid, gBlk, 0, stream>>>(q0_on,  W1, b1, W2, z, b2, onF  + 0, 2);
        vnet_rows_kernel<<<gGrid, gBlk, 0, stream>>>(qtraj,  W1, b1, W2, z, b2, onF  + 1, 2);
    } else {
        // Stage qf_off through the q_trajectory output region, serialized.
        dyn_off_kernel<<<dynGrid, dynBlk, 0, stream>>>(q0_off, p0_off, qtraj, offF);
        vnet_rows_kernel<<<gGrid, gBlk, 0, stream>>>(qtraj,  W1, b1, W2, z, b2, offF + 1, 2);
        vnet_rows_kernel<<<gGrid, gBlk, 0, stream>>>(q0_off, W1, b1, W2, z, b2, offF + 0, 2);
        dyn_on_kernel <<<dynGrid, dynBlk, 0, stream>>>(q0_on, p0_on, u_seq, qtraj, onF);
        vnet_rows_kernel<<<gGrid, gBlk, 0, stream>>>(qtraj,  W1, b1, W2, z, b2, onF + 1, 2);
        vnet_rows_kernel<<<gGrid, gBlk, 0, stream>>>(q0_on,  W1, b1, W2, z, b2, onF + 0, 2);
    }
    (void)in_sizes; (void)n_in; (void)out_size;
}